// BasicSSM_51127290692382
// MI455X (gfx1250) — compile-verified
//
#include <hip/hip_runtime.h>
#include <hip/hip_bf16.h>

typedef float v2f __attribute__((ext_vector_type(2)));
typedef float v8f __attribute__((ext_vector_type(8)));

#define D_MODEL 1024
#define D_STATE 16
#define BATCH   4
#define SEQ     4096
#define CHUNK   64
#define NCHUNK  (SEQ / CHUNK)      // 64
#define NROWS   (BATCH * NCHUNK)   // 256 (b,c) state rows

// ---------------------------------------------------------------------------
// Kernel 1: A_bar = expm(delta*A) via scaling & squaring; also A_bar^CHUNK.
// One block of 256 threads; one thread per matrix element, matrices in LDS.
// ---------------------------------------------------------------------------
__global__ void __launch_bounds__(256)
k_ssm_prep(const float* __restrict__ A, const float* __restrict__ delta,
           float* __restrict__ Abar, float* __restrict__ A64) {
  __shared__ float M[16][16];
  __shared__ float T[16][16];
  const int t = threadIdx.x;       // 0..255
  const int i = t >> 4, j = t & 15;
  const float d = delta[0];

  M[i][j] = A[t] * d * (1.0f / 16.0f);          // scale by 2^-4
  T[i][j] = (i == j) ? 1.0f : 0.0f;
  __syncthreads();

  // Horner Taylor, order 8: T = I + M*T/k
  for (int k = 8; k >= 1; --k) {
    float s = 0.0f;
    #pragma unroll
    for (int m = 0; m < 16; ++m) s += M[i][m] * T[m][j];
    __syncthreads();
    T[i][j] = ((i == j) ? 1.0f : 0.0f) + s * (1.0f / (float)k);
    __syncthreads();
  }
  // square 4x -> expm(delta*A)
  for (int q = 0; q < 4; ++q) {
    float s = 0.0f;
    #pragma unroll
    for (int m = 0; m < 16; ++m) s += T[i][m] * T[m][j];
    __syncthreads();
    T[i][j] = s;
    __syncthreads();
  }
  Abar[t] = T[i][j];
  // 6 more squarings -> (expm(delta*A))^64
  for (int q = 0; q < 6; ++q) {
    float s = 0.0f;
    #pragma unroll
    for (int m = 0; m < 16; ++m) s += T[i][m] * T[m][j];
    __syncthreads();
    T[i][j] = s;
    __syncthreads();
  }
  A64[t] = T[i][j];
}

// ---------------------------------------------------------------------------
// Kernel 2: u = x @ (delta*B)^T  via v_wmma_f32_16x16x4_f32.
// One wave per 16-row tile of x (1024 blocks). K staged through LDS in
// chunks of 64; pad rows to 68 floats (row stride 272B: float4-aligned and
// bank-conflict-free fragment reads: bank = 4*lane + const).
// ---------------------------------------------------------------------------
__global__ void __launch_bounds__(32)
k_ssm_ugemm(const float* __restrict__ x, const float* __restrict__ B,
            const float* __restrict__ delta, float* __restrict__ u) {
  __shared__ alignas(16) float xs[16][68];
  __shared__ alignas(16) float bs[16][68];
  const int lane = threadIdx.x;             // 0..31
  const int m0   = blockIdx.x * 16;         // global row base (b*SEQ + s)
  const float d  = delta[0];
  const int col  = lane & 15;
  const int off  = (lane >> 4) << 1;        // 0 or 2 (K sub-pair)
  v8f acc = {};

  for (int k0 = 0; k0 < D_MODEL; k0 += 64) {
    #pragma unroll
    for (int jj = 0; jj < 8; ++jj) {
      const int q = lane + 32 * jj;         // 0..255 float4 slots
      const int r = q >> 4, c4 = q & 15;
      float4 xv = *(const float4*)(x + (size_t)(m0 + r) * D_MODEL + k0 + c4 * 4);
      *(float4*)&xs[r][c4 * 4] = xv;
      float4 bv = *(const float4*)(B + (size_t)r * D_MODEL + k0 + c4 * 4);
      bv.x *= d; bv.y *= d; bv.z *= d; bv.w *= d;
      *(float4*)&bs[r][c4 * 4] = bv;
    }
    __syncthreads();
    #pragma unroll
    for (int kk = 0; kk < 64; kk += 4) {
      v2f a, b;
      a.x = xs[col][kk + off];
      a.y = xs[col][kk + off + 1];
      b.x = bs[col][kk + off];
      b.y = bs[col][kk + off + 1];
      acc = __builtin_amdgcn_wmma_f32_16x16x4_f32(false, a, false, b,
                                                  (short)0, acc, false, false);
    }
    __syncthreads();
  }
  const int rbase = (lane >> 4) << 3;       // 0 or 8
  #pragma unroll
  for (int i = 0; i < 8; ++i)
    u[(size_t)(m0 + i + rbase) * D_STATE + col] = acc[i];
}

// ---------------------------------------------------------------------------
// Kernel 3: chunk-local scans with zero init; output chunk-final states.
// 128 blocks x 32 threads; each block = 2 (b,c) rows, 16 lanes per row.
// ---------------------------------------------------------------------------
__global__ void __launch_bounds__(32)
k_ssm_scan_local(const float* __restrict__ u, const float* __restrict__ Abar,
                 float* __restrict__ Hloc) {
  __shared__ float hsh[2][16];
  const int lane = threadIdx.x;
  const int half = lane >> 4;
  const int n    = lane & 15;
  const int r    = blockIdx.x * 2 + half;   // 0..255
  const int b    = r >> 6, c = r & 63;

  float arow[16];
  #pragma unroll
  for (int m = 0; m < 16; ++m) arow[m] = Abar[n * 16 + m];

  hsh[half][n] = 0.0f;
  __syncthreads();

  const float* up = u + (size_t)(b * SEQ + c * CHUNK) * D_STATE + n;
  float hn = 0.0f;
  for (int j = 0; j < CHUNK; ++j) {
    float s = up[j * D_STATE];
    #pragma unroll
    for (int m = 0; m < 16; ++m) s += hsh[half][m] * arow[m];
    __syncthreads();
    hsh[half][n] = s;
    __syncthreads();
    hn = s;
  }
  Hloc[r * D_STATE + n] = hn;
}

// ---------------------------------------------------------------------------
// Kernel 4: serial combine across chunks using A_bar^CHUNK.
// One block, 64 threads (4 batches x 16 components). Emits chunk *initial*
// states Hinit.
// ---------------------------------------------------------------------------
__global__ void __launch_bounds__(64)
k_ssm_combine(const float* __restrict__ Hloc, const float* __restrict__ A64,
              float* __restrict__ Hinit) {
  __shared__ float carry[4][16];
  const int t = threadIdx.x;   // 0..63
  const int b = t >> 4, n = t & 15;

  float a64row[16];
  #pragma unroll
  for (int m = 0; m < 16; ++m) a64row[m] = A64[n * 16 + m];

  carry[b][n] = 0.0f;
  __syncthreads();

  for (int c = 0; c < NCHUNK; ++c) {
    Hinit[(b * NCHUNK + c) * D_STATE + n] = carry[b][n];
    float s = Hloc[(b * NCHUNK + c) * D_STATE + n];
    #pragma unroll
    for (int m = 0; m < 16; ++m) s += carry[b][m] * a64row[m];
    __syncthreads();
    carry[b][n] = s;
    __syncthreads();
  }
}

// ---------------------------------------------------------------------------
// Kernel 5: re-scan chunk with correct init (wave0) then y = hs @ C^T with
// WMMA across all 8 waves. 256 blocks (one per (b,c)), 256 threads.
// hs padded to 17 floats/row -> conflict-free A-fragment reads.
// ---------------------------------------------------------------------------
__global__ void __launch_bounds__(256)
k_ssm_scan_ygemm(const float* __restrict__ u, const float* __restrict__ Abar,
                 const float* __restrict__ Hinit, const float* __restrict__ C,
                 float* __restrict__ y) {
  __shared__ alignas(16) float us[CHUNK][16];   // 4 KB u tile
  __shared__ float hs[CHUNK][17];               // padded h tile
  __shared__ float hcur[16];
  const int tid = threadIdx.x;
  const int blk = blockIdx.x;                   // 0..255
  const int b = blk >> 6, c = blk & 63;

  // coalesced u tile load: 1024 floats = 256 float4
  {
    const float* ub = u + (size_t)(b * SEQ + c * CHUNK) * D_STATE;
    float4 v = *(const float4*)(ub + tid * 4);
    *(float4*)&us[0][tid * 4] = v;
  }
  __syncthreads();

  // sequential 64-step scan by 16 lanes of wave 0 (LDS ops in-order per wave)
  if (tid < 16) {
    const int n = tid;
    float arow[16];
    #pragma unroll
    for (int m = 0; m < 16; ++m) arow[m] = Abar[n * 16 + m];
    hcur[n] = Hinit[(b * NCHUNK + c) * D_STATE + n];
    for (int j = 0; j < CHUNK; ++j) {
      float s = us[j][n];
      #pragma unroll
      for (int m = 0; m < 16; ++m) s += hcur[m] * arow[m];
      hcur[n] = s;
      hs[j][n] = s;
    }
  }
  __syncthreads();

  // y tile (64 x 1024) = hs (64x16) @ C^T, split over 8 waves x 128 cols
  const int lane  = tid & 31;
  const int w     = tid >> 5;
  const int col   = lane & 15;
  const int off   = (lane >> 4) << 1;   // 0 or 2
  const int rbase = (lane >> 4) << 3;   // 0 or 8

  for (int nt = 0; nt < 8; ++nt) {
    const int d0 = w * 128 + nt * 16;
    v2f bf[4];
    #pragma unroll
    for (int ks = 0; ks < 4; ++ks) {
      const float* cp = C + (size_t)(d0 + col) * D_STATE + ks * 4 + off;
      bf[ks].x = cp[0];
      bf[ks].y = cp[1];
    }
    #pragma unroll
    for (int mt = 0; mt < 4; ++mt) {
      const int m0 = mt * 16;
      v8f acc = {};
      #pragma unroll
      for (int ks = 0; ks < 4; ++ks) {
        v2f af;
        af.x = hs[m0 + col][ks * 4 + off];
        af.y = hs[m0 + col][ks * 4 + off + 1];
        acc = __builtin_amdgcn_wmma_f32_16x16x4_f32(false, af, false, bf[ks],
                                                    (short)0, acc, false, false);
      }
      float* yp = y + (size_t)(b * SEQ + c * CHUNK + m0 + rbase) * D_MODEL + d0 + col;
      #pragma unroll
      for (int i = 0; i < 8; ++i) yp[(size_t)i * D_MODEL] = acc[i];
    }
  }
}

// ---------------------------------------------------------------------------
// Host launch
// ---------------------------------------------------------------------------
extern "C" void kernel_launch(void* const* d_in, const int* in_sizes, int n_in,
                              void* d_out, int out_size, void* d_ws, size_t ws_size,
                              hipStream_t stream) {
  const float* x     = (const float*)d_in[0];   // (4,4096,1024)
  const float* A     = (const float*)d_in[1];   // (16,16)
  const float* B     = (const float*)d_in[2];   // (16,1024)
  const float* C     = (const float*)d_in[3];   // (1024,16)
  const float* delta = (const float*)d_in[4];   // (1,)
  float* y = (float*)d_out;                     // (4,4096,1024)

  // workspace layout (floats)
  float* ws    = (float*)d_ws;
  float* Abar  = ws;                            // 256
  float* A64   = ws + 256;                      // 256
  float* u     = ws + 512;                      // 4*4096*16 = 262144
  float* Hloc  = u + (size_t)BATCH * SEQ * D_STATE;   // 4096
  float* Hinit = Hloc + NROWS * D_STATE;              // 4096

  k_ssm_prep<<<1, 256, 0, stream>>>(A, delta, Abar, A64);
  k_ssm_ugemm<<<(BATCH * SEQ) / 16, 32, 0, stream>>>(x, B, delta, u);
  k_ssm_scan_local<<<NROWS / 2, 32, 0, stream>>>(u, Abar, Hloc);
  k_ssm_combine<<<1, 64, 0, stream>>>(Hloc, A64, Hinit);
  k_ssm_scan_ygemm<<<NROWS, 256, 0, stream>>>(u, Abar, Hinit, C, y);
}